// fwd_gnn_dense_45174466019868
// MI455X (gfx1250) — compile-verified
//
#include <hip/hip_runtime.h>
#include <hip/hip_bf16.h>

// GNN recurrence over 256 levels. One kernel launch per level (stream-ordered,
// graph-capturable). Each 512-thread block owns 32 nodes. Waves 0-7 execute
// the unary message chain while waves 8-15 execute embed + the binary chain
// concurrently (the two chains are independent until the select), cutting the
// dependent GEMM stages per level from 17 to 12. Matrix math uses
// V_WMMA_F32_16X16X4_F32 (exact fp32 accumulate, matching the reference).

#define L_LVL   256
#define M_NODES 2048
#define FEAT    32
#define H       64

typedef __attribute__((ext_vector_type(2))) float v2f;
typedef __attribute__((ext_vector_type(8))) float v8f;

// Branch-free tanh: tanh(x) = sign(x) * (1-e)/(1+e), e = exp(-2|x|).
// v_exp_f32 + v_rcp_f32, no divergence, saturates to +-1 for large |x|.
__device__ __forceinline__ float fast_tanh(float x) {
  const float ax = __builtin_fabsf(x);
  const float e  = __builtin_amdgcn_exp2f(ax * -2.8853900817779268f); // -2*log2(e)
  const float t  = (1.f - e) * __builtin_amdgcn_rcpf(1.f + e);
  return __builtin_copysignf(t, x);
}

// ---- 16x16 output tile GEMM, K in {32,64,128}, fp32 WMMA -------------------
// A lives in LDS as up to two [32][*] halves (implements concat(left,right)
// along K with zero data movement). W is row-major [K][64] in global (L2-hot).
//
// VGPR layouts per CDNA5 ISA 7.12.2 (32-bit operands):
//   A frag: lane l, elem j -> A[rowBase + (l&15)][k + 2*(l>>4) + j]
//   B frag: lane l, elem j -> W[k + 2*(l>>4) + j][colBase + (l&15)]
//   C/D   : vgpr i, lane l -> D[rowBase + i + 8*(l>>4)][colBase + (l&15)]
template <int K, int LDA>
__device__ __forceinline__ v8f tile_gemm(const float* __restrict__ Al,
                                         const float* __restrict__ Ar,
                                         const float* __restrict__ W,
                                         int rowBase, int colBase, int lane) {
  v8f acc = {0.f, 0.f, 0.f, 0.f, 0.f, 0.f, 0.f, 0.f};
  const int mr = rowBase + (lane & 15);
  const int n  = colBase + (lane & 15);
  const int kh = (lane >> 4) << 1;  // 0 or 2 (uniform across wave halves)
#pragma unroll
  for (int k = 0; k < K; k += 4) {
    const int kk = k + kh;  // k%4==0, kh in {0,2} => (kk<64) == (k<64)
    v2f a, b;
    if (k < 64) {
      a.x = Al[mr * LDA + kk];
      a.y = Al[mr * LDA + kk + 1];
    } else {
      a.x = Ar[mr * H + (kk - 64)];
      a.y = Ar[mr * H + (kk - 63)];
    }
    b.x = W[kk * H + n];
    b.y = W[(kk + 1) * H + n];
    // 8 args: (neg_a, A, neg_b, B, c_mod, C, reuse_a, reuse_b)
    acc = __builtin_amdgcn_wmma_f32_16x16x4_f32(false, a, false, b, (short)0,
                                                acc, false, false);
  }
  return acc;
}

__device__ __forceinline__ void tile_store_tanh(float* __restrict__ dst, int ldD,
                                                v8f acc,
                                                const float* __restrict__ bias,
                                                int rowBase, int colBase, int lane) {
  const int n  = colBase + (lane & 15);
  const float bv = bias[n];
  const int m0 = rowBase + ((lane >> 4) << 3);
#pragma unroll
  for (int i = 0; i < 8; ++i)
    dst[(m0 + i) * ldD + n] = fast_tanh(acc[i] + bv);
}

struct GnnParams {
  const float* feats;   const int* parents; const int* is_unary;
  const float* W_embed; const float* b_embed;
  const float* W_node;  const float* b_node;   // [5][2H][H], [5][H]
  const float* Wu0; const float* bu0; const float* Wu1; const float* bu1;
  const float* Wu;  const float* bu;           // [4][2H][H], [4][H]
  const float* Wb0; const float* bb0; const float* Wb1; const float* bb1;
  const float* Wb;  const float* bb;           // [4][2H][H], [4][H]
  float* out;                                  // [L][M][H]
};

// ---- level 0: out = tanh(feats @ W_embed + b) ------------------------------
__global__ __launch_bounds__(256) void gnn_embed0_kernel(GnnParams p) {
  __shared__ float F[32 * FEAT];
  const int tid = threadIdx.x, lane = tid & 31, wid = tid >> 5;
  const int rowBase = (wid >> 2) * 16, colBase = (wid & 3) * 16;
  const int nodeBase = blockIdx.x * 32;

  for (int idx = tid; idx < 32 * FEAT; idx += 256)
    F[idx] = p.feats[(size_t)(nodeBase + (idx >> 5)) * FEAT + (idx & 31)];
  __syncthreads();

  v8f acc = tile_gemm<FEAT, FEAT>(F, nullptr, p.W_embed, rowBase, colBase, lane);
  tile_store_tanh(p.out + (size_t)nodeBase * H, H, acc, p.b_embed,
                  rowBase, colBase, lane);
}

// ---- one recurrence level ---------------------------------------------------
// Stage schedule (each row = one barrier-separated GEMM stage):
//   s1: [u] mu0 = t(P0@Wu0)        [b] E0 = t(F@W_embed)
//   s2: [u] mu  = t(mu0@Wu1)       [b] mb0 = t([P0|P1]@Wb0)
//   s3: [u] chainU(0)              [b] mb  = t(mb0@Wb1)
//   s4: [u] chainU(1)              [b] chainB(0)
//   s5: [u] chainU(2)              [b] chainB(1)
//   s6: [u] chainU(3)              [b] chainB(2)
//   s7:                            [b] chainB(3)
//   select: RED = is_unary ? TU : TB
//   n1..n5: node fuse chain (waves 0-7), n5 stores to global
__global__ __launch_bounds__(512) void gnn_level_kernel(GnnParams p, int lvl) {
  __shared__ float F[32 * FEAT];
  __shared__ float P0[32 * H], P1[32 * H];
  __shared__ float TU0[32 * H], TU[32 * H];
  __shared__ float TB0[32 * H], TB[32 * H];
  __shared__ float RED[32 * H], E0[32 * H];

  const int tid = threadIdx.x, lane = tid & 31, wid = tid >> 5;
  const int grp = wid >> 3;              // 0 = unary/node, 1 = embed/binary
  const int w   = wid & 7;               // wave within group: 8 tiles/stage
  const int rowBase = (w >> 2) * 16, colBase = (w & 3) * 16;
  const int nodeBase = blockIdx.x * 32;
  const long rowG = (long)lvl * M_NODES + nodeBase;
  const float* outPrev = p.out + (size_t)(lvl - 1) * M_NODES * H;

  // feats tile + parent gather from previous level's results
  for (int idx = tid; idx < 32 * FEAT; idx += 512)
    F[idx] = p.feats[(rowG + (idx >> 5)) * FEAT + (idx & 31)];
  for (int idx = tid; idx < 32 * H; idx += 512) {
    const int r = idx >> 6, h = idx & 63;
    const int pa = p.parents[(rowG + r) * 2 + 0];
    const int pb = p.parents[(rowG + r) * 2 + 1];
    P0[idx] = outPrev[(size_t)pa * H + h];
    P1[idx] = outPrev[(size_t)pb * H + h];
  }
  __syncthreads();

  v8f acc;
  // s1
  if (grp == 0) acc = tile_gemm<H, H>(P0, nullptr, p.Wu0, rowBase, colBase, lane);
  else          acc = tile_gemm<FEAT, FEAT>(F, nullptr, p.W_embed, rowBase, colBase, lane);
  __syncthreads();
  if (grp == 0) tile_store_tanh(TU0, H, acc, p.bu0, rowBase, colBase, lane);
  else          tile_store_tanh(E0, H, acc, p.b_embed, rowBase, colBase, lane);
  __syncthreads();
  // s2
  if (grp == 0) acc = tile_gemm<H, H>(TU0, nullptr, p.Wu1, rowBase, colBase, lane);
  else          acc = tile_gemm<2 * H, H>(P0, P1, p.Wb0, rowBase, colBase, lane);
  __syncthreads();
  if (grp == 0) tile_store_tanh(TU, H, acc, p.bu1, rowBase, colBase, lane);
  else          tile_store_tanh(TB0, H, acc, p.bb0, rowBase, colBase, lane);
  __syncthreads();
  // s3
  if (grp == 0) acc = tile_gemm<2 * H, H>(TU, TU0, p.Wu, rowBase, colBase, lane);
  else          acc = tile_gemm<H, H>(TB0, nullptr, p.Wb1, rowBase, colBase, lane);
  __syncthreads();
  if (grp == 0) tile_store_tanh(TU, H, acc, p.bu, rowBase, colBase, lane);
  else          tile_store_tanh(TB, H, acc, p.bb1, rowBase, colBase, lane);
  __syncthreads();
  // s4..s6: unary chain i=1..3 alongside binary chain i=0..2
#pragma unroll 1
  for (int i = 1; i <= 3; ++i) {
    if (grp == 0) acc = tile_gemm<2 * H, H>(TU, TU0, p.Wu + i * 2 * H * H, rowBase, colBase, lane);
    else          acc = tile_gemm<2 * H, H>(TB, TB0, p.Wb + (i - 1) * 2 * H * H, rowBase, colBase, lane);
    __syncthreads();
    if (grp == 0) tile_store_tanh(TU, H, acc, p.bu + i * H, rowBase, colBase, lane);
    else          tile_store_tanh(TB, H, acc, p.bb + (i - 1) * H, rowBase, colBase, lane);
    __syncthreads();
  }
  // s7: binary chain i=3
  if (grp == 1) acc = tile_gemm<2 * H, H>(TB, TB0, p.Wb + 3 * 2 * H * H, rowBase, colBase, lane);
  __syncthreads();
  if (grp == 1) tile_store_tanh(TB, H, acc, p.bb + 3 * H, rowBase, colBase, lane);
  __syncthreads();

  // select: redux = where(is_unary, unary, binary)
  for (int idx = tid; idx < 32 * H; idx += 512)
    RED[idx] = p.is_unary[rowG + (idx >> 6)] ? TU[idx] : TB[idx];
  __syncthreads();

  // node fuse chain (waves 0-7): e = t([e0|redux]@Wn0), then t([e|e0]@Wn[i])
  if (grp == 0) acc = tile_gemm<2 * H, H>(E0, RED, p.W_node, rowBase, colBase, lane);
  __syncthreads();
  if (grp == 0) tile_store_tanh(TU, H, acc, p.b_node, rowBase, colBase, lane);
  __syncthreads();
#pragma unroll 1
  for (int i = 1; i < 4; ++i) {
    if (grp == 0) acc = tile_gemm<2 * H, H>(TU, E0, p.W_node + i * 2 * H * H, rowBase, colBase, lane);
    __syncthreads();
    if (grp == 0) tile_store_tanh(TU, H, acc, p.b_node + i * H, rowBase, colBase, lane);
    __syncthreads();
  }
  // final stage writes straight to global
  if (grp == 0) {
    acc = tile_gemm<2 * H, H>(TU, E0, p.W_node + 4 * 2 * H * H, rowBase, colBase, lane);
    tile_store_tanh(p.out + (size_t)rowG * H, H, acc, p.b_node + 4 * H,
                    rowBase, colBase, lane);
  }
}

extern "C" void kernel_launch(void* const* d_in, const int* in_sizes, int n_in,
                              void* d_out, int out_size, void* d_ws, size_t ws_size,
                              hipStream_t stream) {
  (void)in_sizes; (void)n_in; (void)out_size; (void)d_ws; (void)ws_size;
  GnnParams p;
  p.feats    = (const float*)d_in[0];
  p.parents  = (const int*)d_in[1];
  p.is_unary = (const int*)d_in[2];
  p.W_embed  = (const float*)d_in[3];
  p.b_embed  = (const float*)d_in[4];
  p.W_node   = (const float*)d_in[5];
  p.b_node   = (const float*)d_in[6];
  p.Wu0 = (const float*)d_in[7];  p.bu0 = (const float*)d_in[8];
  p.Wu1 = (const float*)d_in[9];  p.bu1 = (const float*)d_in[10];
  p.Wu  = (const float*)d_in[11]; p.bu  = (const float*)d_in[12];
  p.Wb0 = (const float*)d_in[13]; p.bb0 = (const float*)d_in[14];
  p.Wb1 = (const float*)d_in[15]; p.bb1 = (const float*)d_in[16];
  p.Wb  = (const float*)d_in[17]; p.bb  = (const float*)d_in[18];
  p.out = (float*)d_out;

  gnn_embed0_kernel<<<dim3(M_NODES / 32), dim3(256), 0, stream>>>(p);
  for (int l = 1; l < L_LVL; ++l)
    gnn_level_kernel<<<dim3(M_NODES / 32), dim3(512), 0, stream>>>(p, l);
}